// RegionLayer_19774029431676
// MI455X (gfx1250) — compile-verified
//
#include <hip/hip_runtime.h>
#include <math.h>

// ---------------- problem constants (fixed by the reference) ----------------
#define NB        64
#define NA        5
#define NC        80
#define NH        38
#define NW        38
#define MAXB      50
#define CH        (5 + NC)        // 85
#define CHTOT     (NA * CH)       // 425
#define SPATIAL   (NH * NW)       // 1444
#define CELLS     (NA * SPATIAL)  // 7220 cells per batch image
#define NTOT      (NB * CELLS)    // 462080
#define BPB       29              // ceil(7220/256) main-kernel blocks per batch
#define MAIN_BLOCKS (NB * BPB)    // 1856
#define PARTN     1920            // partials padded to a multiple of 64 (WMMA chunk)
#define IOU_THRESH 0.6f

__constant__ float c_aw[NA] = {1.3221f, 3.19275f, 5.05587f, 9.47112f, 11.2364f};
__constant__ float c_ah[NA] = {1.73145f, 4.00944f, 8.09892f, 4.84053f, 10.0071f};

typedef float v2f __attribute__((ext_vector_type(2)));
typedef float v8f __attribute__((ext_vector_type(8)));

__device__ __forceinline__ float sigm(float x) { return 1.0f / (1.0f + expf(-x)); }

__device__ __forceinline__ float iou_cxcywh(float ax, float ay, float aw, float ah,
                                            float bx, float by, float bw, float bh) {
    float ax1 = ax - aw * 0.5f, ax2 = ax + aw * 0.5f;
    float ay1 = ay - ah * 0.5f, ay2 = ay + ah * 0.5f;
    float bx1 = bx - bw * 0.5f, bx2 = bx + bw * 0.5f;
    float by1 = by - bh * 0.5f, by2 = by + bh * 0.5f;
    float iw = fmaxf(fminf(ax2, bx2) - fmaxf(ax1, bx1), 0.0f);
    float ih = fmaxf(fminf(ay2, by2) - fmaxf(ay1, by1), 0.0f);
    float inter = iw * ih;
    float uni = aw * ah + bw * bh - inter;
    return inter / fmaxf(uni, 1e-10f);
}

// ---------------- kernel 0: dense defaults + zero partials ----------------
__global__ void k_init(float* __restrict__ obj, float* __restrict__ cmask,
                       float* __restrict__ t0, float* __restrict__ t1,
                       float* __restrict__ t2, float* __restrict__ t3,
                       float* __restrict__ tconf, int* __restrict__ tcls,
                       float* __restrict__ partials) {
    int idx = blockIdx.x * blockDim.x + threadIdx.x;
    if (idx < NTOT) {
        obj[idx]   = 0.0f;
        cmask[idx] = 0.01f;   // SEEN(0) < 12800
        t0[idx]    = 0.5f;    // t0 default
        t1[idx]    = 0.5f;
        t2[idx]    = 0.0f;
        t3[idx]    = 0.0f;
        tconf[idx] = 0.0f;
        tcls[idx]  = 0;
    }
    if (idx < PARTN) partials[idx] = 0.0f;
}

// ---------------- kernel 1: per-batch GT prep + sequential scatter ----------------
__global__ void k_gt(const float* __restrict__ out, const float* __restrict__ target,
                     float* __restrict__ gt, int* __restrict__ nvalid,
                     float* __restrict__ obj, float* __restrict__ cmask,
                     float* __restrict__ t0, float* __restrict__ t1,
                     float* __restrict__ t2, float* __restrict__ t3,
                     float* __restrict__ tconf, int* __restrict__ tcls) {
    int b = blockIdx.x;
    int t = threadIdx.x;

    __shared__ int   s_flag[MAXB];
    __shared__ int   s_nv;
    __shared__ int   s_fidx[MAXB];
    __shared__ float s_cm[MAXB], s_0[MAXB], s_1[MAXB], s_2[MAXB], s_3[MAXB], s_cf[MAXB];
    __shared__ int   s_cl[MAXB];

    float cls = 0.f, x = 0.f, y = 0.f, w = 0.f, h = 0.f;
    if (t < MAXB) {
        const float* tb = target + (size_t)b * (MAXB * 5) + t * 5;
        cls = tb[0]; x = tb[1]; y = tb[2]; w = tb[3]; h = tb[4];
        s_flag[t] = (x > 0.0f) ? 1 : 0;
    }
    __syncthreads();
    if (t == 0) {            // cumprod validity = count until first invalid
        int nv = 0;
        while (nv < MAXB && s_flag[nv]) nv++;
        s_nv = nv;
        nvalid[b] = nv;
    }
    __syncthreads();
    int nv = s_nv;

    if (t < nv) {
        float gx = x * NW, gy = y * NH, gw = w * NW, gh = h * NH;
        gt[((size_t)b * MAXB + t) * 4 + 0] = gx;
        gt[((size_t)b * MAXB + t) * 4 + 1] = gy;
        gt[((size_t)b * MAXB + t) * 4 + 2] = gw;
        gt[((size_t)b * MAXB + t) * 4 + 3] = gh;

        // best anchor (wh-only IoU), argmax keeps first max
        int best = 0; float bestv = -1.0f;
        for (int a = 0; a < NA; ++a) {
            float inter = fminf(c_aw[a], gw) * fminf(c_ah[a], gh);
            float uni   = c_aw[a] * c_ah[a] + gw * gh - inter;
            float r     = inter / fmaxf(uni, 1e-10f);
            if (r > bestv) { bestv = r; best = a; }
        }
        int gi = (int)gx; gi = min(max(gi, 0), NW - 1);
        int gj = (int)gy; gj = min(max(gj, 0), NH - 1);

        // predicted box at the matched cell
        size_t base = ((size_t)(b * CHTOT + best * CH)) * SPATIAL + (size_t)gj * NW + gi;
        float px = sigm(out[base])                 + (float)gi;
        float py = sigm(out[base + SPATIAL])       + (float)gj;
        float pw = expf(out[base + 2 * SPATIAL]) * c_aw[best];
        float ph = expf(out[base + 3 * SPATIAL]) * c_ah[best];

        s_fidx[t] = ((b * NA + best) * NH + gj) * NW + gi;
        s_cm[t]   = 2.0f - w * h;
        s_0[t]    = gx - (float)gi;
        s_1[t]    = gy - (float)gj;
        s_2[t]    = logf(gw / c_aw[best]);
        s_3[t]    = logf(gh / c_ah[best]);
        s_cf[t]   = iou_cxcywh(gx, gy, gw, gh, px, py, pw, ph);
        s_cl[t]   = (int)cls;
    }
    __syncthreads();
    if (t == 0) {            // sequential scatter, fixed order (last write wins)
        for (int q = 0; q < nv; ++q) {
            int f = s_fidx[q];
            obj[f]   = 1.0f;
            cmask[f] = s_cm[q];
            t0[f]    = s_0[q];
            t1[f]    = s_1[q];
            t2[f]    = s_2[q];
            t3[f]    = s_3[q];
            tconf[f] = s_cf[q];
            tcls[f]  = s_cl[q];
        }
    }
}

// ---------------- kernel 2: per-cell loss, deterministic block reduce ----------------
__global__ void k_main(const float* __restrict__ out, const float* __restrict__ gt,
                       const int* __restrict__ nvalid,
                       const float* __restrict__ obj, const float* __restrict__ cmask,
                       const float* __restrict__ t0, const float* __restrict__ t1,
                       const float* __restrict__ t2, const float* __restrict__ t3,
                       const float* __restrict__ tconf, const int* __restrict__ tcls,
                       float* __restrict__ partials) {
    __shared__ float sGT[MAXB * 4];
    __shared__ int   sNV;
    __shared__ float sred[256];

    int b     = blockIdx.x / BPB;
    int local = (blockIdx.x % BPB) * 256 + threadIdx.x;

    if (threadIdx.x < MAXB * 4) sGT[threadIdx.x] = gt[(size_t)b * MAXB * 4 + threadIdx.x];
    if (threadIdx.x == 0)       sNV = nvalid[b];
    __syncthreads();

    float loss = 0.0f;
    if (local < CELLS) {
        int a   = local / SPATIAL;
        int rem = local - a * SPATIAL;
        int j   = rem / NW;
        int i   = rem - j * NW;

        size_t base = ((size_t)(b * CHTOT + a * CH)) * SPATIAL + rem;
        float o0 = out[base];
        float o1 = out[base + SPATIAL];
        float o2 = out[base + 2 * SPATIAL];
        float o3 = out[base + 3 * SPATIAL];
        float o4 = out[base + 4 * SPATIAL];

        float c0 = sigm(o0), c1 = sigm(o1);
        float px = c0 + (float)i, py = c1 + (float)j;
        float pw = expf(o2) * c_aw[a], ph = expf(o3) * c_ah[a];
        float conf = sigm(o4);

        int nv = sNV;
        float mx = 0.0f;
        for (int q = 0; q < nv; ++q) {
            float v = iou_cxcywh(px, py, pw, ph,
                                 sGT[q * 4 + 0], sGT[q * 4 + 1], sGT[q * 4 + 2], sGT[q * 4 + 3]);
            mx = fmaxf(mx, v);
        }

        int n = b * CELLS + local;
        float ob = obj[n];
        float cfm = (ob > 0.0f) ? 5.0f : ((mx <= IOU_THRESH) ? 1.0f : 0.0f);

        float d = (conf - tconf[n]) * cfm;
        loss = d * d;
        float m = cmask[n];
        d = (c0 - t0[n]) * m; loss += d * d;
        d = (c1 - t1[n]) * m; loss += d * d;
        d = (o2 - t2[n]) * m; loss += d * d;
        d = (o3 - t3[n]) * m; loss += d * d;

        if (ob > 0.0f) {      // class CE only at obj cells (sparse)
            size_t cb = base + (size_t)5 * SPATIAL;
            __builtin_prefetch(out + cb, 0, 0);   // global_prefetch_b8
            float mxl = -3.4e38f;
            for (int k = 0; k < NC; ++k) mxl = fmaxf(mxl, out[cb + (size_t)k * SPATIAL]);
            float s = 0.0f;
            for (int k = 0; k < NC; ++k) s += expf(out[cb + (size_t)k * SPATIAL] - mxl);
            float lt = out[cb + (size_t)tcls[n] * SPATIAL];
            loss += (logf(s) + mxl - lt);
        }
    }

    sred[threadIdx.x] = loss;
    __syncthreads();
    for (int s = 128; s > 0; s >>= 1) {
        if (threadIdx.x < s) sred[threadIdx.x] += sred[threadIdx.x + s];
        __syncthreads();
    }
    if (threadIdx.x == 0) partials[blockIdx.x] = sred[0];
}

// ---------------- kernel 3: WMMA-accumulated final reduction ----------------
// One wave (EXEC all-ones as WMMA requires). B = all-ones, so
// D[m][n] = sum_k A[m][k] + C[m][n]: each chunk of 64 fp32 partials is folded
// into the f32 accumulator via V_WMMA_F32_16X16X4_F32, chained through C.
// Every row-sum is replicated across the 16 columns -> exact /16 fixup.
__global__ void k_reduce_wmma(const float* __restrict__ partials, float* __restrict__ outv) {
    int lane = threadIdx.x;          // 0..31
    v8f acc = {};
    v2f bones;
    bones[0] = 1.0f; bones[1] = 1.0f;

    for (int c = 0; c < PARTN / 64; ++c) {
        v2f amat;
        amat[0] = partials[c * 64 + lane];
        amat[1] = partials[c * 64 + 32 + lane];
        acc = __builtin_amdgcn_wmma_f32_16x16x4_f32(
            /*neg_a=*/false, amat, /*neg_b=*/false, bones,
            /*c_mod=*/(short)0, acc, /*reuse_a=*/false, /*reuse_b=*/false);
    }

    float s = acc[0] + acc[1] + acc[2] + acc[3] + acc[4] + acc[5] + acc[6] + acc[7];
    __shared__ float sl[32];
    sl[lane] = s;
    __syncthreads();
    if (lane == 0) {
        float tot = 0.0f;
        for (int k = 0; k < 32; ++k) tot += sl[k];
        outv[0] = tot * (1.0f / 16.0f) * (1.0f / (float)NB);
    }
}

// ---------------- host ----------------
extern "C" void kernel_launch(void* const* d_in, const int* in_sizes, int n_in,
                              void* d_out, int out_size, void* d_ws, size_t ws_size,
                              hipStream_t stream) {
    const float* output = (const float*)d_in[0];
    const float* target = (const float*)d_in[1];

    char* ws = (char*)d_ws;
    size_t off = 0;
    auto alloc = [&](size_t bytes) -> void* {
        void* p = ws + off;
        off += (bytes + 255) & ~(size_t)255;
        return p;
    };
    float* obj      = (float*)alloc((size_t)NTOT * 4);
    float* cmask    = (float*)alloc((size_t)NTOT * 4);
    float* t0       = (float*)alloc((size_t)NTOT * 4);
    float* t1       = (float*)alloc((size_t)NTOT * 4);
    float* t2       = (float*)alloc((size_t)NTOT * 4);
    float* t3       = (float*)alloc((size_t)NTOT * 4);
    float* tconf    = (float*)alloc((size_t)NTOT * 4);
    int*   tcls     = (int*)  alloc((size_t)NTOT * 4);
    float* gt       = (float*)alloc((size_t)NB * MAXB * 4 * 4);
    int*   nvalid   = (int*)  alloc((size_t)NB * 4);
    float* partials = (float*)alloc((size_t)PARTN * 4);
    (void)ws_size; (void)in_sizes; (void)n_in; (void)out_size;

    k_init<<<NTOT / 256, 256, 0, stream>>>(obj, cmask, t0, t1, t2, t3, tconf, tcls, partials);
    k_gt<<<NB, 64, 0, stream>>>(output, target, gt, nvalid,
                                obj, cmask, t0, t1, t2, t3, tconf, tcls);
    k_main<<<MAIN_BLOCKS, 256, 0, stream>>>(output, gt, nvalid,
                                            obj, cmask, t0, t1, t2, t3, tconf, tcls, partials);
    k_reduce_wmma<<<1, 32, 0, stream>>>(partials, (float*)d_out);
}